// LengthRegulator_84997402788025
// MI455X (gfx1250) — compile-verified
//
#include <hip/hip_runtime.h>
#include <stdint.h>

typedef float v4f __attribute__((ext_vector_type(4)));
typedef char __attribute__((address_space(3))) lds_char;

// ---------------------------------------------------------------------------
// Kernel A: per-batch inclusive cumsum of duration (S = blockDim.x = 1024).
// cum[b*S + s] = sum_{i<=s} duration[b,i]; melLen[b] = cum[b*S + S-1].
// ---------------------------------------------------------------------------
__global__ __launch_bounds__(1024) void lr_cumsum_kernel(
    const int* __restrict__ dur, int* __restrict__ cum,
    int* __restrict__ melLen, int S) {
  __shared__ int sh[1024];
  const int b = blockIdx.x;
  const int tid = threadIdx.x;

  sh[tid] = dur[(size_t)b * S + tid];
  __syncthreads();

  // Hillis-Steele inclusive scan
  for (int off = 1; off < 1024; off <<= 1) {
    int v = (tid >= off) ? sh[tid - off] : 0;
    __syncthreads();
    sh[tid] += v;
    __syncthreads();
  }

  cum[(size_t)b * S + tid] = sh[tid];
  if (tid == S - 1) melLen[b] = sh[tid];
}

// ---------------------------------------------------------------------------
// Kernel B: expand rows. Block = 256 threads = 4 rows x 64 lanes.
// Row t of batch b copies x[b, searchsorted_right(cum_b, t), :] (E=256 f32)
// via the CDNA5 async global<->LDS data mover; padded rows store zeros.
// Wave32: each 64-lane row group = 2 full waves -> wave-uniform branches.
// Output stores are non-temporal: the 118 MB output stream is write-once,
// while x (32 MB) is re-read ~3.5x and should stay resident in the 192 MB L2.
// ---------------------------------------------------------------------------
__global__ __launch_bounds__(256) void lr_expand_kernel(
    const float* __restrict__ x, const int* __restrict__ cum,
    const int* __restrict__ melLen, float* __restrict__ out,
    float* __restrict__ mask, int S, int maxMel) {
  constexpr int E = 256;       // floats per row
  constexpr int ROWS = 4;      // rows per block
  __shared__ int s_idx[ROWS];
  __shared__ int s_valid[ROWS];
  __shared__ v4f stage[ROWS * 64];  // 4 KB staging for async copies

  const int b = blockIdx.y;
  const int t0 = blockIdx.x * ROWS;
  const int tid = threadIdx.x;
  const int r = tid >> 6;      // row within block
  const int lane = tid & 63;   // lane within row (each lane moves 16 B)

  if (tid < ROWS) {
    const int t = t0 + tid;
    int valid = 0, idx = 0;
    if (t < maxMel) {
      const int ml = melLen[b];
      valid = (t < ml) ? 1 : 0;
      if (valid) {
        // searchsorted(cum_b, t, side='right'): first idx with cum[idx] > t
        const int* c = cum + (size_t)b * S;
        int lo = 0, hi = S;
        while (lo < hi) {
          int mid = (lo + hi) >> 1;
          if (c[mid] <= t) lo = mid + 1; else hi = mid;
        }
        idx = (lo < S - 1) ? lo : (S - 1);
      }
      mask[(size_t)b * maxMel + t] = valid ? 0.0f : 1.0f;
    }
    s_idx[tid] = idx;
    s_valid[tid] = valid;
  }
  __syncthreads();

  const int t = t0 + r;
  if (t >= maxMel) return;

  const size_t dstElem = ((size_t)b * maxMel + t) * E + (size_t)lane * 4;

  if (s_valid[r]) {
    // Async copy global -> LDS -> global (bypasses VGPRs, uses ASYNCcnt).
    const size_t srcElem = ((size_t)b * S + s_idx[r]) * E + (size_t)lane * 4;
    unsigned goff = (unsigned)(srcElem * sizeof(float));
    unsigned soff = (unsigned)(dstElem * sizeof(float));
    unsigned lds_addr =
        (unsigned)(uintptr_t)(lds_char*)&stage[(r << 6) + lane];

    asm volatile("global_load_async_to_lds_b128 %0, %1, %2"
                 :
                 : "v"(lds_addr), "v"(goff), "s"(x)
                 : "memory");
    asm volatile("s_wait_asynccnt 0" ::: "memory");
    asm volatile("global_store_async_from_lds_b128 %0, %1, %2 th:TH_STORE_NT"
                 :
                 : "v"(soff), "v"(lds_addr), "s"(out)
                 : "memory");
    // s_endpgm performs an implicit wait-idle; stores drain before exit.
  } else {
    v4f z = {0.0f, 0.0f, 0.0f, 0.0f};
    __builtin_nontemporal_store(z, (v4f*)(out + dstElem));
  }
}

// ---------------------------------------------------------------------------
// Host launcher.
// d_out layout: [ out: B*maxMel*E floats | mask: B*maxMel floats (0.0/1.0) ]
// d_ws  layout: [ cum: B*S ints | melLen: B ints ]
// ---------------------------------------------------------------------------
extern "C" void kernel_launch(void* const* d_in, const int* in_sizes, int n_in,
                              void* d_out, int out_size, void* d_ws,
                              size_t ws_size, hipStream_t stream) {
  const float* x = (const float*)d_in[0];
  const int* dur = (const int*)d_in[1];

  const int BS = in_sizes[1];        // B*S = 32768
  const int E = in_sizes[0] / BS;    // 256
  const int B = 32;
  const int S = BS / B;              // 1024
  // out_size = B*maxMel*E + B*maxMel  ->  maxMel
  const int maxMel = out_size / (B * (E + 1));

  int* cum = (int*)d_ws;
  int* melLen = cum + (size_t)B * S;

  float* out = (float*)d_out;
  float* mask = out + (size_t)B * maxMel * E;

  lr_cumsum_kernel<<<dim3(B), dim3(S), 0, stream>>>(dur, cum, melLen, S);

  dim3 grid((maxMel + 3) / 4, B);
  lr_expand_kernel<<<grid, dim3(256), 0, stream>>>(x, cum, melLen, out, mask,
                                                   S, maxMel);
}